// GNNLSTM_90417651516490
// MI455X (gfx1250) — compile-verified
//
#include <hip/hip_runtime.h>

typedef float v2f __attribute__((ext_vector_type(2)));
typedef float v8f __attribute__((ext_vector_type(8)));

#define T_STEPS 8192
#define NNODE   33
#define NEDGE   64
#define DIN     3
#define DH      64
#define DLSTM   128
#define NCLS    5
#define AROWS   48     // 33 rows padded to 3 wmma M-tiles
#define ASTR    132    // 128 + 4 pad: avoids 64-bank stride conflicts
#define BSTR    68     // 64 + 4 pad

__device__ __forceinline__ float sigm(float x) { return 1.0f / (1.0f + __expf(-x)); }
__device__ __forceinline__ float tanh_fast(float x) { return 2.0f / (1.0f + __expf(-2.0f * x)) - 1.0f; }

// ---------------------------------------------------------------------------
// K1: per-timestep fused graph conv1 + conv2 + relu + node-mean -> seq[t][64]
//   conv2 as WMMA GEMM: C(48x64) = A(48x128) @ B(128x64)
//   A = [H1 | AGG2] (rows 33..47 zero), B = [W_root2 ; W_rel2] (col-of-W layout)
// ---------------------------------------------------------------------------
__global__ __launch_bounds__(256) void k_graph_fused(
    const float* __restrict__ x, const int* __restrict__ esrc, const int* __restrict__ edst,
    const float* __restrict__ Wrel1, const float* __restrict__ brel1, const float* __restrict__ Wroot1,
    const float* __restrict__ Wrel2, const float* __restrict__ brel2, const float* __restrict__ Wroot2,
    float* __restrict__ seq)
{
    __shared__ float xs[NNODE * DIN];
    __shared__ float ag1[NNODE * DIN];
    __shared__ float A[AROWS * ASTR];
    __shared__ float Bs[(2 * DH) * BSTR];
    __shared__ float seqacc[DH];
    __shared__ int   es[NEDGE], ed[NEDGE];

    const int tid = threadIdx.x;
    const int t   = blockIdx.x;

    if (tid < NNODE * DIN) xs[tid] = x[t * NNODE * DIN + tid];
    if (tid >= 128 && tid < 128 + NEDGE) {
        es[tid - 128] = esrc[tid - 128];
        ed[tid - 128] = edst[tid - 128];
    }
    if (tid < DH) seqacc[tid] = 0.0f;
    // stage B = [W_root2 ; W_rel2] in K-major layout B[k][n]
    for (int i = tid; i < 2 * DH * DH; i += 256) {
        int k = i >> 6, n = i & 63;
        Bs[k * BSTR + n] = (k < DH) ? Wroot2[n * DH + k] : Wrel2[n * DH + (k - DH)];
    }
    // zero padded A rows 33..47
    for (int i = tid; i < (AROWS - NNODE) * ASTR; i += 256) A[NNODE * ASTR + i] = 0.0f;
    __syncthreads();

    // agg1[n][k] = sum over edges (dst==n) of x[src][k]
    if (tid < NNODE * DIN) {
        int n = tid / DIN, k = tid % DIN;
        float s = 0.0f;
        for (int e = 0; e < NEDGE; ++e)
            if (ed[e] == n) s += xs[es[e] * DIN + k];
        ag1[tid] = s;
    }
    __syncthreads();

    // H1 = agg1 @ Wrel1.T + b + x @ Wroot1.T  -> A[:,0:64]
    for (int i = tid; i < NNODE * DH; i += 256) {
        int n = i >> 6, d = i & 63;
        float v = brel1[d];
#pragma unroll
        for (int k = 0; k < DIN; ++k)
            v += ag1[n * DIN + k] * Wrel1[d * DIN + k] + xs[n * DIN + k] * Wroot1[d * DIN + k];
        A[n * ASTR + d] = v;
    }
    __syncthreads();

    // AGG2[n][d] = sum over edges (dst==n) of H1[src][d]  -> A[:,64:128]
    for (int i = tid; i < NNODE * DH; i += 256) {
        int n = i >> 6, d = i & 63;
        float s = 0.0f;
        for (int e = 0; e < NEDGE; ++e)
            if (ed[e] == n) s += A[es[e] * ASTR + d];
        A[n * ASTR + DH + d] = s;
    }
    __syncthreads();

    // WMMA: 3 x 4 tiles of 16x16, K = 128 in steps of 4 (f32 WMMA)
    const int w = tid >> 5, lane = tid & 31;
    const int l16 = lane & 15, hf = lane >> 4;
    for (int tile = w; tile < 12; tile += 8) {            // wave-uniform trip count
        const int m0 = (tile >> 2) * 16, n0 = (tile & 3) * 16;
        v8f c = {};
        for (int kk = 0; kk < 32; ++kk) {
            const int k0 = kk * 4 + 2 * hf;               // half-wave K split
            v2f a = *(const v2f*)&A[(m0 + l16) * ASTR + k0];
            v2f b;
            b.x = Bs[k0 * BSTR + n0 + l16];
            b.y = Bs[(k0 + 1) * BSTR + n0 + l16];
            c = __builtin_amdgcn_wmma_f32_16x16x4_f32(false, a, false, b, (short)0, c, false, false);
        }
        // epilogue: bias + relu + masked sum over rows -> node mean accumulator
        const int col = n0 + l16;
        const float bias = brel2[col];
        float s = 0.0f;
#pragma unroll
        for (int r = 0; r < 8; ++r) {
            int row = m0 + r + 8 * hf;
            float v = fmaxf(c[r] + bias, 0.0f);
            s += (row < NNODE) ? v : 0.0f;
        }
        atomicAdd(&seqacc[col], s);
    }
    __syncthreads();
    if (tid < DH) seq[t * DH + tid] = seqacc[tid] * (1.0f / 33.0f);
}

// ---------------------------------------------------------------------------
// K2: gates_x(8192,512) = seq(8192,64) @ W_ih.T + (b_ih + b_hh)   (WMMA f32)
// ---------------------------------------------------------------------------
__global__ __launch_bounds__(256) void k_gates_gemm(
    const float* __restrict__ seq, const float* __restrict__ W_ih,
    const float* __restrict__ b_ih, const float* __restrict__ b_hh,
    float* __restrict__ gx)
{
    const int w = threadIdx.x >> 5, lane = threadIdx.x & 31;
    const int l16 = lane & 15, hf = lane >> 4;
    const int id = blockIdx.x * 8 + w;        // 16384 tiles total
    const int m0 = (id >> 5) * 16;            // time rows
    const int n0 = (id & 31) * 16;            // gate cols
    v8f c = {};
    for (int kk = 0; kk < 16; ++kk) {
        const int k0 = kk * 4 + 2 * hf;
        v2f a = *(const v2f*)&seq[(m0 + l16) * DH + k0];
        // B[k][j] = W_ih[j][k]; pairs (k,k+1) contiguous in W_ih row j
        v2f b = *(const v2f*)&W_ih[(n0 + l16) * DH + k0];
        c = __builtin_amdgcn_wmma_f32_16x16x4_f32(false, a, false, b, (short)0, c, false, false);
    }
    const int j = n0 + l16;
    const float bias = b_ih[j] + b_hh[j];
#pragma unroll
    for (int r = 0; r < 8; ++r) {
        int trow = m0 + r + 8 * hf;
        gx[trow * (4 * DLSTM) + j] = c[r] + bias;
    }
}

// ---------------------------------------------------------------------------
// K3: sequential LSTM on one WGP. Thread j owns W_hh row j in registers;
//     h,c,gates live in LDS. Final FC -> out[0..4].
// ---------------------------------------------------------------------------
__global__ __launch_bounds__(512) void k_lstm(
    const float* __restrict__ gx, const float* __restrict__ W_hh,
    const float* __restrict__ W_fc, const float* __restrict__ b_fc,
    float* __restrict__ out)
{
    __shared__ float h[DLSTM], cst[DLSTM], gsm[4 * DLSTM];
    const int j = threadIdx.x;

    float w[DLSTM];
#pragma unroll
    for (int k = 0; k < DLSTM; ++k) w[k] = W_hh[j * DLSTM + k];

    if (j < DLSTM) { h[j] = 0.0f; cst[j] = 0.0f; }
    __syncthreads();

    for (int t = 0; t < T_STEPS; ++t) {
        float g = gx[t * (4 * DLSTM) + j];     // coalesced stream
#pragma unroll
        for (int k = 0; k < DLSTM; ++k) g += w[k] * h[k];  // h[k] LDS broadcast
        gsm[j] = g;
        __syncthreads();
        if (j < DLSTM) {
            float iv = gsm[j], fv = gsm[DLSTM + j], gv = gsm[2 * DLSTM + j], ov = gsm[3 * DLSTM + j];
            float cn = sigm(fv) * cst[j] + sigm(iv) * tanh_fast(gv);
            cst[j] = cn;
            h[j] = sigm(ov) * tanh_fast(cn);
        }
        __syncthreads();
    }

    if (j < NCLS) {
        float s = b_fc[j];
        for (int d = 0; d < DLSTM; ++d) s += W_fc[j * DLSTM + d] * h[d];
        out[j] = s;
    }
}

// ---------------------------------------------------------------------------
extern "C" void kernel_launch(void* const* d_in, const int* in_sizes, int n_in,
                              void* d_out, int out_size, void* d_ws, size_t ws_size,
                              hipStream_t stream)
{
    const float* x      = (const float*)d_in[0];
    const int*   esrc   = (const int*)d_in[1];
    const int*   edst   = (const int*)d_in[2];
    const float* Wrel1  = (const float*)d_in[3];
    const float* brel1  = (const float*)d_in[4];
    const float* Wroot1 = (const float*)d_in[5];
    const float* Wrel2  = (const float*)d_in[6];
    const float* brel2  = (const float*)d_in[7];
    const float* Wroot2 = (const float*)d_in[8];
    const float* W_ih   = (const float*)d_in[9];
    const float* W_hh   = (const float*)d_in[10];
    const float* b_ih   = (const float*)d_in[11];
    const float* b_hh   = (const float*)d_in[12];
    const float* W_fc   = (const float*)d_in[13];
    const float* b_fc   = (const float*)d_in[14];

    float* seq = (float*)d_ws;                 // 8192*64 floats  (2.1 MB)
    float* gx  = seq + (size_t)T_STEPS * DH;   // 8192*512 floats (16.8 MB)
    float* outp = (float*)d_out;

    k_graph_fused<<<T_STEPS, 256, 0, stream>>>(x, esrc, edst, Wrel1, brel1, Wroot1,
                                               Wrel2, brel2, Wroot2, seq);
    k_gates_gemm<<<(T_STEPS / 16) * (4 * DLSTM / 16) / 8, 256, 0, stream>>>(seq, W_ih, b_ih, b_hh, gx);
    k_lstm<<<1, 512, 0, stream>>>(gx, W_hh, W_fc, b_fc, outp);
}